// GNNModel_20117626815225
// MI455X (gfx1250) — compile-verified
//
#include <hip/hip_runtime.h>
#include <hip/hip_bf16.h>

typedef __attribute__((ext_vector_type(2))) float v2f;
typedef __attribute__((ext_vector_type(8))) float v8f;

#define FEAT 128
#define LDS_STRIDE 132   // 132 mod 64 = 4 -> conflict-free paired b64 reads

__device__ __forceinline__ void atomic_add_f32(float* p, float v) {
    __hip_atomic_fetch_add(p, v, __ATOMIC_RELAXED, __HIP_MEMORY_SCOPE_AGENT);
}

// ---------------- generic fill ----------------
__global__ __launch_bounds__(256) void fill_kernel(float* p, float val, int n) {
    int gid = blockIdx.x * 256 + threadIdx.x;
    if (gid < n) p[gid] = val;
}

// ---------------- degree accumulation (edges only; self-loop pre-seeded) ----
__global__ __launch_bounds__(256) void deg_kernel(const int* __restrict__ dst,
                                                  float* deg, int E) {
    int e = blockIdx.x * 256 + threadIdx.x;
    if (e < E) atomic_add_f32(&deg[dst[e]], 1.0f);
}

__global__ __launch_bounds__(256) void rsqrt_kernel(float* d, int n) {
    int gid = blockIdx.x * 256 + threadIdx.x;
    if (gid < n) d[gid] = rsqrtf(d[gid]);
}

// ---------------- WMMA fp32 GEMM: Y[N,128] = X[N,128] @ W[128,128] ----------
// Block: 256 threads = 8 waves; each wave does 16 rows x 128 cols.
// Uses V_WMMA_F32_16X16X4_F32, K-loop of 32 steps.
__global__ __launch_bounds__(256)
void gemm128_kernel(const float* __restrict__ X, const float* __restrict__ W,
                    float* __restrict__ Y, int nrows) {
    __shared__ float lw[FEAT * LDS_STRIDE];  // W transposed: lw[n*S + k]
    __shared__ float la[FEAT * LDS_STRIDE];  // A tile:       la[r*S + k]

    const int tid  = threadIdx.x;
    const int lane = tid & 31;
    const int wave = tid >> 5;
    const int block_row = blockIdx.x * 128;

    // Stage W (row-major [k][n]) transposed into LDS
    for (int i4 = tid; i4 < (FEAT * FEAT) / 4; i4 += 256) {
        int k  = i4 >> 5;            // 32 float4 per row of 128
        int n0 = (i4 & 31) << 2;
        float4 w4 = *(const float4*)&W[k * FEAT + n0];
        lw[(n0 + 0) * LDS_STRIDE + k] = w4.x;
        lw[(n0 + 1) * LDS_STRIDE + k] = w4.y;
        lw[(n0 + 2) * LDS_STRIDE + k] = w4.z;
        lw[(n0 + 3) * LDS_STRIDE + k] = w4.w;
    }
    // Stage A tile (128 rows) with coalesced float4 loads, zero-pad OOB rows
    for (int i4 = tid; i4 < (128 * FEAT) / 4; i4 += 256) {
        int r  = i4 >> 5;
        int k0 = (i4 & 31) << 2;
        int row = block_row + r;
        float4 xv = make_float4(0.f, 0.f, 0.f, 0.f);
        if (row < nrows) xv = *(const float4*)&X[(size_t)row * FEAT + k0];
        *(float4*)&la[r * LDS_STRIDE + k0] = xv;
    }
    __syncthreads();

    const int m0    = wave * 16;
    const int arow  = m0 + (lane & 15);
    const int khalf = (lane >> 4) << 1;   // lanes 16-31 carry K+2
    const int ncol  = lane & 15;

    v8f acc[8];
    const v8f zero = {0.f, 0.f, 0.f, 0.f, 0.f, 0.f, 0.f, 0.f};
#pragma unroll
    for (int t = 0; t < 8; ++t) acc[t] = zero;

    for (int k0 = 0; k0 < FEAT; k0 += 4) {
        int ak = k0 + khalf;
        // A frag: 16x4 f32 -> lane holds (K=ak, K=ak+1) of row arow
        float2 av = *(const float2*)&la[arow * LDS_STRIDE + ak];
        v2f a; a.x = av.x; a.y = av.y;
#pragma unroll
        for (int t = 0; t < 8; ++t) {
            // B frag: 4x16 f32 -> lane holds (K=ak, K=ak+1) of col n
            float2 bv = *(const float2*)&lw[(t * 16 + ncol) * LDS_STRIDE + ak];
            v2f b; b.x = bv.x; b.y = bv.y;
            acc[t] = __builtin_amdgcn_wmma_f32_16x16x4_f32(
                false, a, false, b, (short)0, acc[t], false, false);
        }
    }

    // Store C/D: VGPR j -> rows m0+j (lanes 0-15) / m0+j+8 (lanes 16-31)
    const int rbase = block_row + m0 + ((lane >> 4) << 3);
#pragma unroll
    for (int t = 0; t < 8; ++t) {
#pragma unroll
        for (int j = 0; j < 8; ++j) {
            int row = rbase + j;
            if (row < nrows)
                Y[(size_t)row * FEAT + t * 16 + (lane & 15)] = acc[t][j];
        }
    }
}

// ---------------- edge scatter: acc[dst] += h[src] * dis[src]*dis[dst] ------
// One wave per edge; each lane handles 4 features (float4 gather + 4 atomics)
__global__ __launch_bounds__(256)
void scatter_kernel(const float* __restrict__ h, const int* __restrict__ src,
                    const int* __restrict__ dst, const float* __restrict__ dis,
                    float* acc, int E) {
    int gid  = blockIdx.x * 256 + threadIdx.x;
    int e    = gid >> 5;
    int lane = gid & 31;
    if (e >= E) return;
    int s = src[e], d = dst[e];
    float nrm = dis[s] * dis[d];
    float4 v = *(const float4*)&h[(size_t)s * FEAT + lane * 4];
    float* base = &acc[(size_t)d * FEAT + lane * 4];
    atomic_add_f32(base + 0, v.x * nrm);
    atomic_add_f32(base + 1, v.y * nrm);
    atomic_add_f32(base + 2, v.z * nrm);
    atomic_add_f32(base + 3, v.w * nrm);
}

// ---------------- self-loop + bias + relu -----------------------------------
__global__ __launch_bounds__(256)
void sbr_kernel(const float* __restrict__ acc, const float* __restrict__ h,
                const float* __restrict__ dis, const float* __restrict__ bias,
                float* out, int total) {
    int gid = blockIdx.x * 256 + threadIdx.x;
    if (gid >= total) return;
    int i = gid >> 7;
    int f = gid & 127;
    float dd = dis[i];
    float v = acc[gid] + h[gid] * dd * dd + bias[f];
    out[gid] = v > 0.f ? v : 0.f;
}

// ---------------- mean-pool accumulation ------------------------------------
__global__ __launch_bounds__(256)
void pool_kernel(const float* __restrict__ h, const int* __restrict__ batch,
                 float* sums, float* cnts, int N) {
    int gid  = blockIdx.x * 256 + threadIdx.x;
    int i    = gid >> 5;
    int lane = gid & 31;
    if (i >= N) return;
    int g = batch[i];
    float4 v = *(const float4*)&h[(size_t)i * FEAT + lane * 4];
    float* base = &sums[(size_t)g * FEAT + lane * 4];
    atomic_add_f32(base + 0, v.x);
    atomic_add_f32(base + 1, v.y);
    atomic_add_f32(base + 2, v.z);
    atomic_add_f32(base + 3, v.w);
    if (lane == 0) atomic_add_f32(&cnts[g], 1.0f);
}

// ---------------- head: (sums/cnt) @ Wfc + bfc ------------------------------
__global__ __launch_bounds__(256)
void final_kernel(const float* __restrict__ sums, const float* __restrict__ cnts,
                  const float* __restrict__ Wfc, const float* __restrict__ bfc,
                  float* out, int G) {
    int g = blockIdx.x * 256 + threadIdx.x;
    if (g >= G) return;
    float inv = 1.0f / fmaxf(cnts[g], 1.0f);
    float a = 0.f;
    for (int f = 0; f < FEAT; ++f) a += sums[(size_t)g * FEAT + f] * Wfc[f];
    out[g] = a * inv + bfc[0];
}

static inline int blk(long long n) { return (int)((n + 255) / 256); }

extern "C" void kernel_launch(void* const* d_in, const int* in_sizes, int n_in,
                              void* d_out, int out_size, void* d_ws, size_t ws_size,
                              hipStream_t stream) {
    const float* x    = (const float*)d_in[0];
    const int*   edge = (const int*)d_in[1];
    const int*   batch= (const int*)d_in[2];
    const float* W1   = (const float*)d_in[3];
    const float* b1   = (const float*)d_in[4];
    const float* W2   = (const float*)d_in[5];
    const float* b2   = (const float*)d_in[6];
    const float* Wfc  = (const float*)d_in[7];
    const float* bfc  = (const float*)d_in[8];

    const int N = in_sizes[0] / FEAT;
    const int E = in_sizes[1] / 2;
    const int G = out_size;           // pooled output is [G,1]
    const int* srcIdx = edge;
    const int* dstIdx = edge + E;

    // workspace carve-up (floats)
    float* dis  = (float*)d_ws;                       // N (deg -> dis in place)
    float* buf0 = dis + (((size_t)N + 255) & ~(size_t)255);
    float* buf1 = buf0 + (size_t)N * FEAT;
    float* sums = buf1 + (size_t)N * FEAT;
    float* cnts = sums + (size_t)G * FEAT;

    const long long NF = (long long)N * FEAT;

    // degree (self-loop seeds deg=1) -> dis = rsqrt(deg)
    fill_kernel<<<blk(N), 256, 0, stream>>>(dis, 1.0f, N);
    deg_kernel<<<blk(E), 256, 0, stream>>>(dstIdx, dis, E);
    rsqrt_kernel<<<blk(N), 256, 0, stream>>>(dis, N);

    // ---- layer 1 ----
    gemm128_kernel<<<(N + 127) / 128, 256, 0, stream>>>(x, W1, buf0, N);  // hx
    fill_kernel<<<blk(NF), 256, 0, stream>>>(buf1, 0.0f, (int)NF);        // acc
    scatter_kernel<<<blk((long long)E * 32), 256, 0, stream>>>(buf0, srcIdx, dstIdx, dis, buf1, E);
    sbr_kernel<<<blk(NF), 256, 0, stream>>>(buf1, buf0, dis, b1, buf0, (int)NF); // h1 -> buf0

    // ---- layer 2 ----
    gemm128_kernel<<<(N + 127) / 128, 256, 0, stream>>>(buf0, W2, buf1, N); // hx2 -> buf1
    fill_kernel<<<blk(NF), 256, 0, stream>>>(buf0, 0.0f, (int)NF);          // acc2 -> buf0
    scatter_kernel<<<blk((long long)E * 32), 256, 0, stream>>>(buf1, srcIdx, dstIdx, dis, buf0, E);
    sbr_kernel<<<blk(NF), 256, 0, stream>>>(buf0, buf1, dis, b2, buf1, (int)NF); // h2 -> buf1

    // ---- pool + head ----
    fill_kernel<<<blk((long long)G * FEAT), 256, 0, stream>>>(sums, 0.0f, G * FEAT);
    fill_kernel<<<blk(G), 256, 0, stream>>>(cnts, 0.0f, G);
    pool_kernel<<<blk((long long)N * 32), 256, 0, stream>>>(buf1, batch, sums, cnts, N);
    final_kernel<<<blk(G), 256, 0, stream>>>(sums, cnts, Wfc, bfc, (float*)d_out, G);
}